// ROIRelationLayers_50225347559758
// MI455X (gfx1250) — compile-verified
//
#include <hip/hip_runtime.h>
#include <hip/hip_bf16.h>
#include <math.h>

// ---------------------------------------------------------------------------
// ROIRelationLayers on MI455X (gfx1250): fp32 WMMA (16x16x4) everywhere.
// B=2, N=512, D=256, H=8, dh=32, L=3.
// ---------------------------------------------------------------------------

#define BB 2
#define NN 512
#define DD 256
#define HH 8
#define LL 3
#define DHD 32            // head dim
#define EPSF 1e-4f

typedef __attribute__((ext_vector_type(2))) float v2f;
typedef __attribute__((ext_vector_type(8))) float v8f;

static __device__ __forceinline__ v8f wmma_f32_4(v2f a, v2f b, v8f c) {
  // V_WMMA_F32_16X16X4_F32 : D = A(16x4) * B(4x16) + C(16x16), all fp32
  return __builtin_amdgcn_wmma_f32_16x16x4_f32(false, a, false, b, (short)0, c,
                                               false, false);
}

// ---------------------------------------------------------------------------
// Kernel 1: fused positional-embedding -> Wg projection -> log(relu(.)+1e-6)
// One thread per (b, i, j) pair; all L layers / H heads at once.
// glog layout: [l][b][h][i][j]
// ---------------------------------------------------------------------------
__global__ __launch_bounds__(256) void geo_kernel(
    const float* __restrict__ boxes, const float* __restrict__ Wg,
    const float* __restrict__ bg, const float* __restrict__ divm,
    float* __restrict__ glog) {
  __shared__ float wgl[LL * DD * HH];   // 6144 floats = 24 KB
  __shared__ float bgl[LL * HH];
  __shared__ float invdiv[DD / 8];

  const int tid = threadIdx.x;
  for (int t = tid; t < LL * DD * HH; t += 256) wgl[t] = Wg[t];
  if (tid < LL * HH) bgl[tid] = bg[tid];
  if (tid < DD / 8) invdiv[tid] = 1.0f / divm[tid];
  __syncthreads();

  const size_t idx = (size_t)blockIdx.x * 256 + tid;
  const int j = (int)(idx & (NN - 1));
  const int i = (int)((idx >> 9) & (NN - 1));
  const int b = (int)(idx >> 18);

  const float* bi = boxes + ((size_t)b * NN + i) * 4;
  const float* bj = boxes + ((size_t)b * NN + j) * 4;
  const float cxi = 0.5f * (bi[0] + bi[2]), cyi = 0.5f * (bi[1] + bi[3]);
  const float wi = fmaxf(bi[2] - bi[0], EPSF), hi = fmaxf(bi[3] - bi[1], EPSF);
  const float cxj = 0.5f * (bj[0] + bj[2]), cyj = 0.5f * (bj[1] + bj[3]);
  const float wj = fmaxf(bj[2] - bj[0], EPSF), hj = fmaxf(bj[3] - bj[1], EPSF);

  // i = query row, j = key column (matches reference broadcasting)
  float rel[4];
  rel[0] = logf(fabsf(cxj - cxi) / wj + EPSF);
  rel[1] = logf(fabsf(cyj - cyi) / hj + EPSF);
  rel[2] = logf(wi / wj + EPSF);
  rel[3] = logf(hi / hj + EPSF);

  float acc[LL * HH];
#pragma unroll
  for (int t = 0; t < LL * HH; ++t) acc[t] = 0.0f;

  for (int g = 0; g < 4; ++g) {
    const float rg = rel[g];
    for (int t = 0; t < 32; ++t) {
      float s, c;
      sincosf(rg * invdiv[t], &s, &c);
      const int ds = g * 64 + t;      // sin feature index
      const int dc = ds + 32;         // cos feature index
#pragma unroll
      for (int lh = 0; lh < LL * HH; ++lh) {
        const int l = lh >> 3, h = lh & 7;
        acc[lh] = fmaf(s, wgl[(l * DD + ds) * HH + h], acc[lh]);
        acc[lh] = fmaf(c, wgl[(l * DD + dc) * HH + h], acc[lh]);
      }
    }
  }

#pragma unroll
  for (int lh = 0; lh < LL * HH; ++lh) {
    const int l = lh >> 3, h = lh & 7;
    const float g = fmaxf(acc[lh] + bgl[l * HH + h], 0.0f);
    glog[((((size_t)l * BB + b) * HH + h) * NN + i) * NN + j] =
        logf(g + 1e-6f);
  }
}

// ---------------------------------------------------------------------------
// Kernel 2: OUT = relu(A[Mx256] * W[256x256] + bias) (+ residual, templated)
// 4 waves / block share one column tile tn; W panel staged transposed in LDS
// (stride 261 -> conflict-free b64 reads). Inner loop:
//   1x global_load_b64 (A) + 1x ds_load_b64 (B) + 1x v_wmma_f32_16x16x4_f32.
// ---------------------------------------------------------------------------
#define WSTR 261  // padded LDS stride for transposed W panel (floats)

template <bool RES>
__global__ __launch_bounds__(128) void gemm_relu16(
    const float* __restrict__ A, const float* __restrict__ W,
    const float* __restrict__ bias, const float* __restrict__ residual,
    float* __restrict__ out, int M) {
  __shared__ float wp[16 * WSTR];    // W panel, transposed: wp[col][k]

  const int wave = threadIdx.x >> 5;
  const int lane = threadIdx.x & 31;
  const int tn = blockIdx.x & 15;              // DD/16 = 16 column tiles
  const int tm = (blockIdx.x >> 4) * 4 + wave; // 4 row tiles per block

  // cooperative transposed stage of W[:, tn*16 : tn*16+16]
  for (int t = threadIdx.x; t < 16 * DD; t += 128) {
    const int col = t & 15;          // 0..15  within tile
    const int kk = t >> 4;           // 0..255 reduction index
    wp[col * WSTR + kk] = W[(size_t)kk * DD + tn * 16 + col];
  }
  __syncthreads();

  if (tm * 16 < M) {                 // wave-uniform; EXEC all-ones inside
    const int m = lane & 15;         // row (A) / col (B,C)
    const int hh = lane >> 4;        // k-half selector
    const float* arow = A + ((size_t)(tm * 16 + m)) * DD + 2 * hh;
    const float* wrow = &wp[m * WSTR + 2 * hh];

    v8f c = {};
    for (int k = 0; k < DD; k += 4) {
      v2f a = *(const v2f*)(arow + k);
      v2f b = *(const v2f*)(wrow + k);   // {W[k+2hh][col], W[k+2hh+1][col]}
      c = wmma_f32_4(a, b, c);
    }

    const float bval = bias[tn * 16 + m];
#pragma unroll
    for (int v = 0; v < 8; ++v) {
      const int row = tm * 16 + v + 8 * hh;
      const int col = tn * 16 + m;
      float val = fmaxf(c[v] + bval, 0.0f);
      if (RES) val += residual[(size_t)row * DD + col];
      out[(size_t)row * DD + col] = val;
    }
  }
}

// ---------------------------------------------------------------------------
// Kernel 3: attention for one layer. One wave per (b, h, 16-query tile).
// scores tile (16 x 512) lives in LDS; exact softmax; P*V via WMMA.
// ---------------------------------------------------------------------------
#define SSTR 516  // padded LDS row stride (floats)

__global__ __launch_bounds__(32) void attn16(
    const float* __restrict__ q, const float* __restrict__ k,
    const float* __restrict__ v, const float* __restrict__ glog,
    const int* __restrict__ mask, float* __restrict__ y, int layer) {
  __shared__ float S[16 * SSTR];
  __shared__ float rowinv[16];

  const int lane = threadIdx.x;
  const int bid = blockIdx.x;        // b*H*32 + h*32 + qt
  const int qt = bid & 31;
  const int h = (bid >> 5) & 7;
  const int b = bid >> 8;
  const int qbase = qt * 16;
  const int m = lane & 15;
  const int hh = lane >> 4;
  const float scale = 0.17677669529663687f;  // 1/sqrt(32)

  // Q tile A-operands, kept in registers (16x32 fp32 across the wave)
  v2f aq[8];
  {
    const float* qrow = q + ((size_t)(b * NN + qbase + m)) * DD + h * DHD;
#pragma unroll
    for (int kk = 0; kk < 8; ++kk) aq[kk] = *(const v2f*)(qrow + 4 * kk + 2 * hh);
  }

  const size_t glbase = (((size_t)layer * BB + b) * HH + h) * NN;  // *NN later
  const size_t mbase = (size_t)b * NN;
  const float* kcol = k + ((size_t)(b * NN + m)) * DD + h * DHD;

  // ---- scores = Q*K^T * scale + glog, masked; stream K tiles from L2 ----
  for (int kt = 0; kt < 32; ++kt) {
    const int kbase = kt * 16;
    const float* krow = kcol + (size_t)kbase * DD;
    // prefetch next K tile (wrapped so the last iteration stays in-bounds)
    __builtin_prefetch(kcol + (size_t)((kbase + 16) & (NN - 1)) * DD, 0, 0);
    v8f c = {};
#pragma unroll
    for (int kk = 0; kk < 8; ++kk) {
      v2f bkv = *(const v2f*)(krow + 4 * kk + 2 * hh);
      c = wmma_f32_4(aq[kk], bkv, c);
    }
#pragma unroll
    for (int vv = 0; vv < 8; ++vv) {
      const int row = vv + 8 * hh;       // query within tile
      const int col = kbase + m;         // key index
      float sc = c[vv] * scale +
                 glog[(glbase + (qbase + row)) * NN + col];
      if (mask[(mbase + qbase + row) * NN + col] == 0) sc = -1e9f;
      S[row * SSTR + col] = sc;
    }
  }
  __syncthreads();

  // ---- exact softmax over 512 keys: 2 lanes per row ----
  {
    const int r = lane >> 1;
    const int part = lane & 1;
    float mx = -3.4e38f;
    for (int cix = part * 256; cix < part * 256 + 256; ++cix)
      mx = fmaxf(mx, S[r * SSTR + cix]);
    mx = fmaxf(mx, __shfl_xor(mx, 1, 32));
    float sum = 0.0f;
    for (int cix = part * 256; cix < part * 256 + 256; ++cix) {
      const float e = __expf(S[r * SSTR + cix] - mx);
      S[r * SSTR + cix] = e;
      sum += e;
    }
    sum += __shfl_xor(sum, 1, 32);
    if (part == 0) rowinv[r] = 1.0f / sum;
  }
  __syncthreads();

  // ---- y = (P * V) * rowinv ; A from LDS, V streamed from L2 ----
  v8f c0 = {}, c1 = {};
  const float* vb = v + (size_t)b * NN * DD + h * DHD;
  for (int kk4 = 0; kk4 < NN; kk4 += 4) {
    v2f a = *(const v2f*)(&S[m * SSTR + kk4 + 2 * hh]);
    const int k0 = kk4 + 2 * hh;
    v2f b0, b1;
    b0.x = vb[(size_t)k0 * DD + m];
    b0.y = vb[(size_t)(k0 + 1) * DD + m];
    b1.x = vb[(size_t)k0 * DD + 16 + m];
    b1.y = vb[(size_t)(k0 + 1) * DD + 16 + m];
    c0 = wmma_f32_4(a, b0, c0);
    c1 = wmma_f32_4(a, b1, c1);
  }
#pragma unroll
  for (int vv = 0; vv < 8; ++vv) {
    const int row = vv + 8 * hh;
    const float inv = rowinv[row];
    const size_t orow = ((size_t)(b * NN + qbase + row)) * DD + h * DHD;
    y[orow + m] = c0[vv] * inv;
    y[orow + 16 + m] = c1[vv] * inv;
  }
}

// ---------------------------------------------------------------------------
__global__ __launch_bounds__(256) void copy_f32x4(const float4* __restrict__ src,
                                                  float4* __restrict__ dst,
                                                  int n4) {
  const int i = blockIdx.x * 256 + threadIdx.x;
  if (i < n4) dst[i] = src[i];
}

// ---------------------------------------------------------------------------
extern "C" void kernel_launch(void* const* d_in, const int* in_sizes, int n_in,
                              void* d_out, int out_size, void* d_ws,
                              size_t ws_size, hipStream_t stream) {
  const float* boxes = (const float*)d_in[0];
  const float* features = (const float*)d_in[1];
  const int* mask = (const int*)d_in[2];
  const float* Wq = (const float*)d_in[3];
  const float* bq = (const float*)d_in[4];
  const float* Wk = (const float*)d_in[5];
  const float* bk = (const float*)d_in[6];
  const float* Wv = (const float*)d_in[7];
  const float* bv = (const float*)d_in[8];
  const float* Wo = (const float*)d_in[9];
  const float* bo = (const float*)d_in[10];
  const float* Wg = (const float*)d_in[11];
  const float* bg = (const float*)d_in[12];
  const float* divm = (const float*)d_in[13];

  float* x = (float*)d_out;                    // running activations (B,N,D)
  float* ws = (float*)d_ws;
  const size_t glog_sz = (size_t)LL * BB * HH * NN * NN;  // 12.6M floats
  const size_t xd = (size_t)BB * NN * DD;                 // 262144
  float* glog = ws;
  float* qbuf = glog + glog_sz;
  float* kbuf = qbuf + xd;
  float* vbuf = kbuf + xd;
  float* ybuf = vbuf + xd;

  // 1) fused geometry embedding -> log-geo for all layers/heads (once)
  geo_kernel<<<(BB * NN * NN) / 256, 256, 0, stream>>>(boxes, Wg, bg, divm,
                                                       glog);
  // 2) x = features
  copy_f32x4<<<(int)(xd / 4 / 256), 256, 0, stream>>>(
      (const float4*)features, (float4*)x, (int)(xd / 4));

  const int M = BB * NN;                               // 1024 rows
  const int gemm_blocks = ((M / 16 + 3) / 4) * (DD / 16);  // 16*16 = 256

  for (int l = 0; l < LL; ++l) {
    const size_t wo = (size_t)l * DD * DD;
    const size_t bo_ = (size_t)l * DD;
    gemm_relu16<false><<<gemm_blocks, 128, 0, stream>>>(x, Wq + wo, bq + bo_,
                                                        nullptr, qbuf, M);
    gemm_relu16<false><<<gemm_blocks, 128, 0, stream>>>(x, Wk + wo, bk + bo_,
                                                        nullptr, kbuf, M);
    gemm_relu16<false><<<gemm_blocks, 128, 0, stream>>>(x, Wv + wo, bv + bo_,
                                                        nullptr, vbuf, M);
    attn16<<<BB * HH * (NN / 16), 32, 0, stream>>>(qbuf, kbuf, vbuf, glog,
                                                   mask, ybuf, l);
    gemm_relu16<true><<<gemm_blocks, 128, 0, stream>>>(ybuf, Wo + wo, bo + bo_,
                                                       x, x, M);
  }
}